// SAT_84670985274022
// MI455X (gfx1250) — compile-verified
//
#include <hip/hip_runtime.h>
#include <hip/hip_bf16.h>
#include <math.h>

// Problem constants (match reference)
#define B_   64
#define N_   4096
#define D2_  128
#define S_   512

typedef __attribute__((ext_vector_type(2))) float v2f;
typedef __attribute__((ext_vector_type(8))) float v8f;

// ---------------------------------------------------------------------------
// Kernel 1: q[b, e] = sum_d yq[b, d] * w_mat[d, e]   (64x128 = 64x128 @ 128x128)
// fp32 WMMA: V_WMMA_F32_16X16X4_F32, one wave per 16x16 output tile.
// ISA layouts (CDNA5 ISA §7.12.2):
//   A 16x4 : lanes 0-15 -> M=lane,  VGPR0=K0, VGPR1=K1 ; lanes 16-31 -> K2,K3
//   B 4x16 : lanes 0-15 -> N=lane,  VGPR0=K0, VGPR1=K1 ; lanes 16-31 -> K2,K3
//   C 16x16: VGPR r -> M=r (lanes 0-15), M=r+8 (lanes 16-31), N=lane&15
// ---------------------------------------------------------------------------
__global__ __launch_bounds__(32) void SAT_k_qproj(const float* __restrict__ yq,
                                                  const float* __restrict__ w,
                                                  float* __restrict__ q) {
    const int lane = threadIdx.x;        // 0..31 (wave32)
    const int mt   = blockIdx.x & 3;     // 4 M-tiles (B_/16)
    const int nt   = blockIdx.x >> 2;    // 8 N-tiles (D2_/16)
    const int half = lane >> 4;          // 0: lanes 0-15, 1: lanes 16-31
    const int l15  = lane & 15;
    const int m    = mt * 16 + l15;      // yq row (= batch index)
    const int ncol = nt * 16 + l15;      // w_mat column

    v8f c = {};
    #pragma unroll 4
    for (int k0 = 0; k0 < D2_; k0 += 4) {
        const int ka = k0 + half * 2;
        v2f a, bm;
        a.x  = yq[m * D2_ + ka];
        a.y  = yq[m * D2_ + ka + 1];
        bm.x = w[ka * D2_ + ncol];
        bm.y = w[(ka + 1) * D2_ + ncol];
        // 8-arg form: (neg_a, A, neg_b, B, c_mod, C, reuse_a, reuse_b)
        c = __builtin_amdgcn_wmma_f32_16x16x4_f32(false, a, false, bm,
                                                  (short)0, c, false, false);
    }

    #pragma unroll
    for (int r = 0; r < 8; ++r) {
        q[(mt * 16 + r + half * 8) * D2_ + ncol] = c[r];
    }
}

// ---------------------------------------------------------------------------
// Kernel 2: logits[b, n] = q[b, :] . y_past[b, n, :]
// 1024 blocks (64 b x 16 chunks of 256 n), 256 threads = 8 waves.
// Each wave computes one row-dot per step: lane l holds q[b, 4l..4l+3] and
// loads y_past[b, n, 4l..4l+3] -> one coalesced 512B load per wave, then a
// 5-step shfl_xor tree reduction (wave32).
// ---------------------------------------------------------------------------
__global__ __launch_bounds__(256) void SAT_k_logits(const float* __restrict__ y_past,
                                                    const float* __restrict__ q,
                                                    float* __restrict__ logits) {
    const int b     = blockIdx.x >> 4;
    const int chunk = blockIdx.x & 15;
    const int tid   = threadIdx.x;
    const int lane  = tid & 31;
    const int wave  = tid >> 5;

    const float4 q4 = ((const float4*)(q + b * D2_))[lane];
    const float4* __restrict__ ybase =
        (const float4*)(y_past + (size_t)b * N_ * D2_);
    float* __restrict__ lout = logits + (size_t)b * N_;

    const int n0 = chunk * 256 + wave * 32;
    for (int i = 0; i < 32; ++i) {
        const int n = n0 + i;
        const float4 y4 = ybase[(size_t)n * (D2_ / 4) + lane];
        float p = fmaf(y4.x, q4.x, fmaf(y4.y, q4.y, fmaf(y4.z, q4.z, y4.w * q4.w)));
        #pragma unroll
        for (int off = 16; off; off >>= 1) p += __shfl_xor(p, off, 32);
        if (lane == 0) lout[n] = p;
    }
}

// ---------------------------------------------------------------------------
// Kernel 3: per-batch softmax over N_ logits + scatter-add by s_past into a
// 512-bin LDS histogram (no global atomics), then coalesced row store.
// One block per batch, 256 threads = 8 waves.
// ---------------------------------------------------------------------------
__global__ __launch_bounds__(256) void SAT_k_softmax_scatter(
        const float* __restrict__ logits,
        const int*   __restrict__ s_past,
        float*       __restrict__ out) {
    __shared__ float e[N_];      // 16 KB
    __shared__ float post[S_];   // 2 KB
    __shared__ float red[9];

    const int b    = blockIdx.x;
    const int tid  = threadIdx.x;
    const int lane = tid & 31;
    const int wave = tid >> 5;

    const float* __restrict__ lrow = logits + (size_t)b * N_;

    for (int s = tid; s < S_; s += 256) post[s] = 0.f;

    // Pass 1: load logits into LDS, block max
    float lmax = -INFINITY;
    for (int n = tid; n < N_; n += 256) {
        const float v = lrow[n];
        e[n] = v;
        lmax = fmaxf(lmax, v);
    }
    #pragma unroll
    for (int off = 16; off; off >>= 1) lmax = fmaxf(lmax, __shfl_xor(lmax, off, 32));
    if (lane == 0) red[wave] = lmax;
    __syncthreads();
    if (tid == 0) {
        float m = red[0];
        for (int i = 1; i < 8; ++i) m = fmaxf(m, red[i]);
        red[8] = m;
    }
    __syncthreads();
    const float gmax = red[8];

    // Pass 2: exp and block sum
    float lsum = 0.f;
    for (int n = tid; n < N_; n += 256) {
        const float ev = expf(e[n] - gmax);
        e[n] = ev;
        lsum += ev;
    }
    #pragma unroll
    for (int off = 16; off; off >>= 1) lsum += __shfl_xor(lsum, off, 32);
    __syncthreads();
    if (lane == 0) red[wave] = lsum;
    __syncthreads();
    if (tid == 0) {
        float s = 0.f;
        for (int i = 0; i < 8; ++i) s += red[i];
        red[8] = s;
    }
    __syncthreads();
    const float inv = 1.0f / red[8];

    // Pass 3: scatter-add into LDS histogram
    const int* __restrict__ srow = s_past + (size_t)b * N_;
    for (int n = tid; n < N_; n += 256) {
        atomicAdd(&post[srow[n]], e[n] * inv);
    }
    __syncthreads();

    // Coalesced store of the output row (fully overwrites poison)
    for (int s = tid; s < S_; s += 256) out[b * S_ + s] = post[s];
}

// ---------------------------------------------------------------------------
// Launch: inputs in setup_inputs() order: s_past(i32), yq(f32), y_past(f32),
// w_mat(f32), size_s(scalar, compile-time 512 here). Output: (B, S) f32.
// Workspace: q (32 KB) + logits (1 MB).
// ---------------------------------------------------------------------------
extern "C" void kernel_launch(void* const* d_in, const int* in_sizes, int n_in,
                              void* d_out, int out_size, void* d_ws, size_t ws_size,
                              hipStream_t stream) {
    const int*   s_past = (const int*)d_in[0];
    const float* yq     = (const float*)d_in[1];
    const float* y_past = (const float*)d_in[2];
    const float* w_mat  = (const float*)d_in[3];
    float*       out    = (float*)d_out;

    float* q      = (float*)d_ws;                                  // 64*128 f32
    float* logits = (float*)((char*)d_ws + (size_t)B_ * D2_ * sizeof(float)); // 64*4096 f32

    SAT_k_qproj<<<32, 32, 0, stream>>>(yq, w_mat, q);
    SAT_k_logits<<<B_ * 16, 256, 0, stream>>>(y_past, q, logits);
    SAT_k_softmax_scatter<<<B_, 256, 0, stream>>>(logits, s_past, out);
}